// FFTDepthwiseConv_13091060318283
// MI455X (gfx1250) — compile-verified
//
#include <hip/hip_runtime.h>
#include <cstdint>

// Problem constants (from reference): B=8, H=W=224, C=192, K=7 circular depthwise conv.
#define BB 8
#define HH 224
#define WW 224
#define CC 192
#define KK 7
#define RAD (KK / 2)

// Tile: 8 out-rows x 16 out-cols x 32 channels per 256-thread block.
#define TH 8
#define TW 16
#define TC 32
#define XR   (TH + KK - 1)  // 14 LDS rows
#define XCOL (TW + KK - 1)  // 22 LDS cols

// ---- CDNA5 async global->LDS path (ASYNCcnt-tracked), via inline asm -------
// CDNA5 ISA: GLOBAL_LOAD_ASYNC_TO_LDS_B128
//   VDST = per-lane LDS byte offset (wave-relative), VADDR = 64-bit global
//   address, SADDR = off. Tracked by ASYNCcnt; wave waits with s_wait_asynccnt.
__device__ __forceinline__ void stage16(const float* __restrict__ g, float* l) {
#if defined(__HIP_DEVICE_COMPILE__)
    // Low 32 bits of a generic LDS pointer == wave-relative LDS byte offset
    // (ISA §10.2 aperture rule).
    uint32_t lds_off = (uint32_t)(uintptr_t)l;
    asm volatile("global_load_async_to_lds_b128 %0, %1, off"
                 :
                 : "v"(lds_off), "v"(g)
                 : "memory");
#else
    *(float4*)l = *(const float4*)g;   // host-pass placeholder (never codegen'd)
#endif
}

__device__ __forceinline__ void wait_async_lds() {
#if defined(__HIP_DEVICE_COMPILE__)
    asm volatile("s_wait_asynccnt 0" ::: "memory");
#endif
}
// ---------------------------------------------------------------------------

#define FMA4(a, xv, wv)                      \
    do {                                     \
        (a).x = fmaf((xv).x, (wv).x, (a).x); \
        (a).y = fmaf((xv).y, (wv).y, (a).y); \
        (a).z = fmaf((xv).z, (wv).z, (a).z); \
        (a).w = fmaf((xv).w, (wv).w, (a).w); \
    } while (0)

__global__ __launch_bounds__(256) void dwconv7_circ_kernel(
    const float* __restrict__ x,     // (B,H,W,C)
    const float* __restrict__ ker,   // (C,K,K)
    float* __restrict__ out)         // (B,H,W,C)
{
    __shared__ __align__(16) float xs[XR * XCOL * TC];   // 39424 B
    __shared__ __align__(16) float wk[KK * KK * TC];     //  6272 B

    const int tid = threadIdx.x;
    const int w0  = blockIdx.x * TW;
    const int h0  = blockIdx.y * TH;
    const int zc  = blockIdx.z;               // 0..47
    const int b   = zc / (CC / TC);
    const int c0  = (zc % (CC / TC)) * TC;

    // ---- Stage haloed input tile into LDS, circular wrap in addresses ----
    // Total float4 chunks = 14*22*32/4 = 2464; LDS float index == 4*t.
    for (int t = tid; t < XR * XCOL * TC / 4; t += 256) {
        const int ch4 = t & (TC / 4 - 1);     // 0..7
        const int pix = t >> 3;
        const int col = pix % XCOL;
        const int row = pix / XCOL;
        int gh = h0 + row - RAD; if (gh < 0) gh += HH; if (gh >= HH) gh -= HH;
        int gw = w0 + col - RAD; if (gw < 0) gw += WW; if (gw >= WW) gw -= WW;
        const size_t goff = (((size_t)b * HH + gh) * WW + gw) * CC + c0 + ch4 * 4;
        stage16(x + goff, &xs[(size_t)t * 4]);
    }

    // ---- Stage weights, pre-flipped: wk[(ky,kx),cc] = ker[c0+cc, 6-ky, 6-kx] ----
    for (int t = tid; t < KK * KK * TC; t += 256) {
        const int cc  = t & (TC - 1);
        const int tap = t >> 5;               // 0..48
        const int ky  = tap / KK, kx = tap % KK;
        wk[tap * TC + cc] =
            ker[(size_t)(c0 + cc) * (KK * KK) + (KK - 1 - ky) * KK + (KK - 1 - kx)];
    }

    wait_async_lds();
    __syncthreads();

    // ---- Compute: thread = (4 channels) x (1 col) x (4 consecutive rows) ----
    // Sliding 4-deep register window along ky cuts xs LDS loads 196 -> 70/thread.
    const int ch  = (tid & 7) * 4;            // channel offset within tile
    const int col = (tid >> 3) & (TW - 1);    // output col within tile
    const int q   = tid >> 7;                 // 0 or 1
    const int r0  = q * 4;                    // first output row of this thread

    float4 acc0 = make_float4(0.f, 0.f, 0.f, 0.f);
    float4 acc1 = acc0, acc2 = acc0, acc3 = acc0;

    for (int kx = 0; kx < KK; ++kx) {
        const int cb = (col + kx) * TC + ch;
        // window holds input rows r0+ky .. r0+ky+3
        float4 win0 = *(const float4*)&xs[(r0 + 0) * (XCOL * TC) + cb];
        float4 win1 = *(const float4*)&xs[(r0 + 1) * (XCOL * TC) + cb];
        float4 win2 = *(const float4*)&xs[(r0 + 2) * (XCOL * TC) + cb];
        float4 win3 = *(const float4*)&xs[(r0 + 3) * (XCOL * TC) + cb];
#pragma unroll
        for (int ky = 0; ky < KK; ++ky) {
            const float4 wv = *(const float4*)&wk[(ky * KK + kx) * TC + ch];
            FMA4(acc0, win0, wv);
            FMA4(acc1, win1, wv);
            FMA4(acc2, win2, wv);
            FMA4(acc3, win3, wv);
            if (ky < KK - 1) {   // shift window down one row (regs rename via unroll)
                win0 = win1; win1 = win2; win2 = win3;
                win3 = *(const float4*)&xs[(r0 + ky + 4) * (XCOL * TC) + cb];
            }
        }
    }

    // ---- Store (coalesced b128) ----
    const size_t obase =
        (((size_t)b * HH + (h0 + r0)) * WW + (w0 + col)) * CC + c0 + ch;
    *(float4*)&out[obase + 0 * (size_t)(WW * CC)] = acc0;
    *(float4*)&out[obase + 1 * (size_t)(WW * CC)] = acc1;
    *(float4*)&out[obase + 2 * (size_t)(WW * CC)] = acc2;
    *(float4*)&out[obase + 3 * (size_t)(WW * CC)] = acc3;
}

extern "C" void kernel_launch(void* const* d_in, const int* in_sizes, int n_in,
                              void* d_out, int out_size, void* d_ws, size_t ws_size,
                              hipStream_t stream) {
    (void)in_sizes; (void)n_in; (void)out_size; (void)d_ws; (void)ws_size;
    const float* x   = (const float*)d_in[0];   // (8,224,224,192) f32
    const float* ker = (const float*)d_in[1];   // (192,7,7) f32
    float* out = (float*)d_out;                 // (8,224,224,192) f32

    dim3 grid(WW / TW, HH / TH, BB * (CC / TC));   // (14, 28, 48)
    dwconv7_circ_kernel<<<grid, 256, 0, stream>>>(x, ker, out);
}